// Classify_node_12919261627062
// MI455X (gfx1250) — compile-verified
//
#include <hip/hip_runtime.h>
#include <math.h>

typedef float v2f __attribute__((ext_vector_type(2)));
typedef float v8f __attribute__((ext_vector_type(8)));

#define HID 128

// ---------------- degree / dinv ----------------
__global__ void deg_init_kernel(float* __restrict__ deg, int n) {
  int i = blockIdx.x * blockDim.x + threadIdx.x;
  if (i < n) deg[i] = 1.0f;  // self loop
}

__global__ void deg_accum_kernel(const int* __restrict__ dst, float* __restrict__ deg, int e) {
  int i = blockIdx.x * blockDim.x + threadIdx.x;
  if (i < e) atomicAdd(&deg[dst[i]], 1.0f);
}

__global__ void deg_rsqrt_kernel(float* __restrict__ deg, int n) {
  int i = blockIdx.x * blockDim.x + threadIdx.x;
  if (i < n) deg[i] = rsqrtf(deg[i]);  // deg >= 1 always (self loops)
}

// ---------------- pad Wl (128x40) -> Bpad (128x48, zero-filled) ----------------
__global__ void pad_weights_kernel(const float* __restrict__ Wl, float* __restrict__ Bpad) {
  int i = blockIdx.x * blockDim.x + threadIdx.x;
  if (i < HID * 48) {
    int r = i / 48, c = i % 48;
    Bpad[i] = (c < 40) ? Wl[r * 40 + c] : 0.0f;
  }
}

// ---------------- WMMA GEMM: C[M x nvalid] = A[M x 128] * B[128 x NCOLS] (+bias) ----
// NCOLS/LDC compile-time; B zero-padded beyond nvalid.
// grid.x = M/32; blockDim.x = 32*(NCOLS/16); one wave per n-tile, 2 m-tiles/wave.
// All A/B accesses index directly off the __restrict__ kernel args (keeps them in
// addrspace(1) -> global_load with scalar base + immediate offsets, no flat/DScnt).
// ISA 7.12.2 32-bit layouts:
//   A 16x4:  lanes 0-15 -> M=0..15; V0=K(0|2), V1=K(1|3) by lane half.
//   B 4x16:  lanes 0-15 hold K=0..1 across V0..V1; lanes 16-31 hold K=2..3.
//   C 16x16: VGPR v: lanes 0-15 -> M=v, lanes 16-31 -> M=v+8; N = lane%16.
template <int NCOLS, int LDC>
__global__ void __launch_bounds__(256)
gemm_wmma_f32(const float* __restrict__ A, const float* __restrict__ B,
              const float* __restrict__ bias, float* __restrict__ C,
              int nvalid) {
  const int lane  = threadIdx.x & 31;
  const int tn    = threadIdx.x >> 5;
  const int m0    = blockIdx.x * 32;
  const int rlane = lane & 15;
  const int khalf = (lane >> 4) << 1;  // 0 or 2
  const int n     = tn * 16 + rlane;

  const int aOff0 = (m0 + rlane) * HID + khalf;        // m-tile 0
  const int aOff1 = aOff0 + 16 * HID;                  // m-tile 1
  const int bOff  = khalf * NCOLS + n;

  v8f acc0 = {}, acc1 = {};

#pragma unroll 4
  for (int kk = 0; kk < 32; ++kk) {
    v2f b;
    b.x = B[bOff + kk * 4 * NCOLS];
    b.y = B[bOff + kk * 4 * NCOLS + NCOLS];
    v2f a0 = *(const v2f*)(A + aOff0 + kk * 4);
    v2f a1 = *(const v2f*)(A + aOff1 + kk * 4);
    acc0 = __builtin_amdgcn_wmma_f32_16x16x4_f32(false, a0, false, b,
                                                 (short)0, acc0, false, false);
    acc1 = __builtin_amdgcn_wmma_f32_16x16x4_f32(false, a1, false, b,
                                                 (short)0, acc1, false, false);
  }

  if (n < nvalid) {
    const float bv = bias ? bias[n] : 0.0f;
    const int cBase = (m0 + (lane >> 4) * 8) * LDC + n;
#pragma unroll
    for (int v = 0; v < 8; ++v)
      C[cBase + v * LDC] = acc0[v] + bv;
#pragma unroll
    for (int v = 0; v < 8; ++v)
      C[cBase + (16 + v) * LDC] = acc1[v] + bv;
  }
}

// ---------------- self-loop init: agg[i] = dinv[i]^2 * tmp[i] ----------------
__global__ void self_init_kernel(const float* __restrict__ tmp, const float* __restrict__ dinv,
                                 float* __restrict__ agg, long total) {
  long i = (long)blockIdx.x * blockDim.x + threadIdx.x;
  if (i < total) {
    float c = dinv[i >> 7];  // node = i / 128
    agg[i] = c * c * tmp[i];
  }
}

// ---------------- edge scatter: agg[dst] += dinv[src]*dinv[dst]*tmp[src] ------
// One wave (32 lanes) per edge; each lane gathers float4 and does 4 f32 atomics.
__global__ void __launch_bounds__(256)
edge_scatter_kernel(const int* __restrict__ src, const int* __restrict__ dst,
                    const float* __restrict__ dinv, const float* __restrict__ tmp,
                    float* __restrict__ agg, int e) {
  const int wave = blockIdx.x * (blockDim.x >> 5) + (threadIdx.x >> 5);
  const int lane = threadIdx.x & 31;
  if (wave >= e) return;
  const int s = src[wave];
  const int d = dst[wave];
  const float c = dinv[s] * dinv[d];
  const float4 v = ((const float4*)(tmp + (size_t)s * HID))[lane];
  float* drow = agg + (size_t)d * HID + lane * 4;
  atomicAdd(drow + 0, c * v.x);
  atomicAdd(drow + 1, c * v.y);
  atomicAdd(drow + 2, c * v.z);
  atomicAdd(drow + 3, c * v.w);
}

// ---------------- bias + ELU (in place) ----------------
__global__ void bias_elu_kernel(float* __restrict__ h, const float* __restrict__ b, long total) {
  long i = (long)blockIdx.x * blockDim.x + threadIdx.x;
  if (i < total) {
    float x = h[i] + b[i & (HID - 1)];
    h[i] = (x > 0.0f) ? x : expm1f(x);
  }
}

// ---------------- log_softmax over 40 classes, one wave per node ----------------
__global__ void __launch_bounds__(256)
logsoftmax_kernel(const float* __restrict__ logits, float* __restrict__ out, int n) {
  const int node = blockIdx.x * (blockDim.x >> 5) + (threadIdx.x >> 5);
  const int lane = threadIdx.x & 31;
  if (node >= n) return;
  const float* r = logits + (size_t)node * 40;
  const bool hi = (lane + 32) < 40;
  float x0 = r[lane];
  float x1 = hi ? r[lane + 32] : -3.402823466e38f;
  float m = fmaxf(x0, x1);
#pragma unroll
  for (int off = 16; off > 0; off >>= 1) m = fmaxf(m, __shfl_xor(m, off, 32));
  float s = __expf(x0 - m) + (hi ? __expf(x1 - m) : 0.0f);
#pragma unroll
  for (int off = 16; off > 0; off >>= 1) s += __shfl_xor(s, off, 32);
  const float lse = m + __logf(s);
  float* o = out + (size_t)node * 40;
  o[lane] = x0 - lse;
  if (hi) o[lane + 32] = x1 - lse;
}

// ---------------- launcher ----------------
extern "C" void kernel_launch(void* const* d_in, const int* in_sizes, int n_in,
                              void* d_out, int out_size, void* d_ws, size_t ws_size,
                              hipStream_t stream) {
  const float* x  = (const float*)d_in[0];
  const int* ei   = (const int*)d_in[1];
  const float* W0 = (const float*)d_in[2];
  const float* b0 = (const float*)d_in[3];
  const float* W1 = (const float*)d_in[4];
  const float* b1 = (const float*)d_in[5];
  const float* W2 = (const float*)d_in[6];
  const float* b2 = (const float*)d_in[7];
  const float* Wl = (const float*)d_in[8];
  const float* bl = (const float*)d_in[9];
  float* out = (float*)d_out;

  const int N = in_sizes[0] / HID;   // 100000 (multiple of 32)
  const int E = in_sizes[1] / 2;     // 3200000
  const int NC = 40;

  char* ws = (char*)d_ws;
  const size_t szDinv = ((size_t)N * sizeof(float) + 255) & ~(size_t)255;
  const size_t szBuf  = ((size_t)N * HID * sizeof(float) + 255) & ~(size_t)255;
  float* dinv = (float*)ws;
  float* bufA = (float*)(ws + szDinv);                    // agg / h
  float* bufB = (float*)(ws + szDinv + szBuf);            // tmp = h*W, then logits
  float* Bpad = (float*)(ws + szDinv + 2 * szBuf);        // 128x48 padded Wl

  const int* srcIdx = ei;
  const int* dstIdx = ei + E;

  const long totalNH = (long)N * HID;
  const int TB = 256;

  // degrees -> dinv
  deg_init_kernel<<<(N + TB - 1) / TB, TB, 0, stream>>>(dinv, N);
  deg_accum_kernel<<<(E + TB - 1) / TB, TB, 0, stream>>>(dstIdx, dinv, E);
  deg_rsqrt_kernel<<<(N + TB - 1) / TB, TB, 0, stream>>>(dinv, N);

  // zero-pad classifier weights 128x40 -> 128x48
  pad_weights_kernel<<<(HID * 48 + TB - 1) / TB, TB, 0, stream>>>(Wl, Bpad);

  const int edgeBlocks = (E + (TB / 32) - 1) / (TB / 32);
  const int nhBlocks   = (int)((totalNH + TB - 1) / TB);
  const int gemmBlocks = N / 32;  // 32 rows per block (2 m-tiles per wave)

  const float* h = x;
  const float* Ws[3] = {W0, W1, W2};
  const float* bs[3] = {b0, b1, b2};
  for (int l = 0; l < 3; ++l) {
    // tmp = h @ W  (8 n-tiles * 32 lanes = 256 threads)
    gemm_wmma_f32<HID, HID><<<gemmBlocks, 8 * 32, 0, stream>>>(h, Ws[l], nullptr, bufB, HID);
    // agg = dinv^2 * tmp (self loops)
    self_init_kernel<<<nhBlocks, TB, 0, stream>>>(bufB, dinv, bufA, totalNH);
    // agg[dst] += dinv[src]*dinv[dst]*tmp[src]
    edge_scatter_kernel<<<edgeBlocks, TB, 0, stream>>>(srcIdx, dstIdx, dinv, bufB, bufA, E);
    // h = elu(agg + b)
    bias_elu_kernel<<<nhBlocks, TB, 0, stream>>>(bufA, bs[l], totalNH);
    h = bufA;
  }

  // logits = h @ Bpad + bl  (48 padded cols -> 3 n-tiles; store/bias guarded to 40)
  gemm_wmma_f32<48, 40><<<gemmBlocks, 3 * 32, 0, stream>>>(h, Bpad, bl, bufB, NC);
  // log_softmax rows -> d_out
  logsoftmax_kernel<<<(N + (TB / 32) - 1) / (TB / 32), TB, 0, stream>>>(bufB, out, N);
}